// QuantumLatentLayer_65481071407766
// MI455X (gfx1250) — compile-verified
//
#include <hip/hip_runtime.h>

typedef __attribute__((ext_vector_type(2))) float v2f;
typedef __attribute__((ext_vector_type(8))) float v8f;

#define BATCH   32768
#define LATENT  2048
#define NQ      24
#define NPAD    32

// Stage W (24 x 2048) into a zero-padded 32 x 2048 copy in workspace so the
// GEMM inner loop needs no column masking and B loads are uniform.
__global__ __launch_bounds__(256) void pad_w_kernel(const float* __restrict__ W,
                                                    float* __restrict__ Wp) {
    int idx = blockIdx.x * 256 + threadIdx.x;      // 0 .. 32*2048-1
    int row = idx >> 11;                           // idx / 2048
    float v = 0.0f;
    if (row < NQ) v = W[idx];
    Wp[idx] = v;
}

__global__ __launch_bounds__(256) void qlatent_gemm_kernel(
    const float* __restrict__ latent,   // [BATCH][LATENT]
    const float* __restrict__ Wp,       // [NPAD][LATENT] (rows >= 24 are zero)
    const float* __restrict__ bias,     // [NQ]
    const float* __restrict__ params,   // [NQ][3]
    float* __restrict__ out)            // [BATCH][NQ]
{
    const int lane = threadIdx.x & 31;
    const int wave = threadIdx.x >> 5;
    const int lo16 = lane & 15;   // A row-in-tile / B-C-D column-in-tile
    const int hi   = lane >> 4;   // selects K pair (A/B) and row half (C/D)

    // Each wave computes a 32-row x 32-col (24 valid) output tile.
    const int m0 = (blockIdx.x * 16 + wave * 2) * 16;   // block covers 256 rows
    const int m1 = m0 + 16;

    // Per-lane base pointers. 16x4 f32 A tile: VGPR0={K0|K2}, VGPR1={K1|K3}
    // => per lane a contiguous float2 at column k + 2*hi.
    const float* aptr0 = latent + (size_t)(m0 + lo16) * LATENT + 2 * hi;
    const float* aptr1 = latent + (size_t)(m1 + lo16) * LATENT + 2 * hi;
    // 4x16 f32 B tile: B[K][N] = Wp[N][K]; same contiguous float2 pattern.
    const float* bptr0 = Wp + (size_t)lo16 * LATENT + 2 * hi;
    const float* bptr1 = Wp + (size_t)(lo16 + 16) * LATENT + 2 * hi;

    v8f c00 = {}; v8f c01 = {}; v8f c10 = {}; v8f c11 = {};

#pragma unroll 8
    for (int k = 0; k < LATENT; k += 4) {
        // latent is a one-pass 256MB stream: non-temporal so W stays in L2.
        v2f a0 = __builtin_nontemporal_load((const v2f*)(aptr0 + k));
        v2f a1 = __builtin_nontemporal_load((const v2f*)(aptr1 + k));
        v2f b0 = *(const v2f*)(bptr0 + k);
        v2f b1 = *(const v2f*)(bptr1 + k);
        c00 = __builtin_amdgcn_wmma_f32_16x16x4_f32(false, a0, false, b0,
                                                    (short)0, c00, false, false);
        c01 = __builtin_amdgcn_wmma_f32_16x16x4_f32(false, a0, false, b1,
                                                    (short)0, c01, false, false);
        c10 = __builtin_amdgcn_wmma_f32_16x16x4_f32(false, a1, false, b0,
                                                    (short)0, c10, false, false);
        c11 = __builtin_amdgcn_wmma_f32_16x16x4_f32(false, a1, false, b1,
                                                    (short)0, c11, false, false);
    }

    // Epilogue: per-lane qubit index is fixed per N-tile.
    const int q0 = lo16;            // always < 24
    const int q1 = 16 + lo16;       // valid only when lo16 < 8
    const bool q1v = (lo16 < 8);

    float bq0  = bias[q0];
    float phi0 = params[q0 * 3 + 0];
    float th0  = params[q0 * 3 + 1];
    float ct0  = __cosf(th0);
    float cs0  = __cosf(phi0) * __sinf(th0);

    float bq1 = 0.f, ct1 = 0.f, cs1 = 0.f;
    if (q1v) {
        bq1 = bias[q1];
        float phi1 = params[q1 * 3 + 0];
        float th1  = params[q1 * 3 + 1];
        ct1 = __cosf(th1);
        cs1 = __cosf(phi1) * __sinf(th1);
    }

#pragma unroll
    for (int i = 0; i < 8; ++i) {
        const int r0 = m0 + i + 8 * hi;   // C/D row mapping: VGPR i, half 'hi'
        const int r1 = m1 + i + 8 * hi;

        float a  = c00[i] + bq0;
        out[(size_t)r0 * NQ + q0] = __cosf(a) * ct0 - __sinf(a) * cs0;
        float a3 = c10[i] + bq0;
        out[(size_t)r1 * NQ + q0] = __cosf(a3) * ct0 - __sinf(a3) * cs0;
        if (q1v) {
            float a2 = c01[i] + bq1;
            out[(size_t)r0 * NQ + q1] = __cosf(a2) * ct1 - __sinf(a2) * cs1;
            float a4 = c11[i] + bq1;
            out[(size_t)r1 * NQ + q1] = __cosf(a4) * ct1 - __sinf(a4) * cs1;
        }
    }
}

extern "C" void kernel_launch(void* const* d_in, const int* in_sizes, int n_in,
                              void* d_out, int out_size, void* d_ws, size_t ws_size,
                              hipStream_t stream) {
    const float* latent = (const float*)d_in[0];   // [32768][2048] f32
    const float* W      = (const float*)d_in[1];   // [24][2048] f32
    const float* bias   = (const float*)d_in[2];   // [24] f32
    const float* params = (const float*)d_in[3];   // [24][3] f32
    float* out = (float*)d_out;                    // [32768][24] f32
    float* Wp  = (float*)d_ws;                     // 32*2048*4 = 256 KB scratch

    pad_w_kernel<<<(NPAD * LATENT) / 256, 256, 0, stream>>>(W, Wp);
    qlatent_gemm_kernel<<<BATCH / 256, 256, 0, stream>>>(latent, Wp, bias,
                                                         params, out);
}